// TBEInputPrepareReference_12472585028199
// MI455X (gfx1250) — compile-verified
//
#include <hip/hip_runtime.h>
#include <hip/hip_bf16.h>

// ---------------------------------------------------------------------------
// TBE input-prepare for MI455X (gfx1250): pure bandwidth problem (~268 MB of
// traffic -> ~11.5us floor at 23.3 TB/s). No matrix math -> no WMMA; the
// relevant CDNA5 feature is the async global<->LDS copy path (ASYNCcnt):
//   - aligned segments (indices): async b128 loads + async b128 stores,
//     wave-local LDS slices, no barriers at all.
//   - misaligned segments (weights; dest starts at an odd word because the
//     combined-offsets section has 8*8192+1 entries): async b128 loads +
//     dword async stores from LDS, which absorb the +1-word shift while
//     staying fully coalesced (128 contiguous bytes per wave instruction).
// Data never touches VGPRs; NT temporal hints keep the one-touch stream from
// rinsing the 192 MB L2.
// ---------------------------------------------------------------------------

#define TILE_CHUNKS 512   // 512 x 16B = 8 KiB LDS tile = 2048 dwords
#define TPB 256           // 8 waves; 2 chunks per thread per tile

struct SegParams {
  const int* src[16];   // 0..7 indices tables, 8..15 weights tables
  long long  dst[16];   // destination word offset in out
  int        cnt[16];   // word count
};

struct OffParams {
  const int* off[8];
  long long  dst[8];    // destination word offset of table's offset block
  int        bcnt[8];   // entries taken per table (= len-1, include_last_offset)
  int        rebase[8]; // accumulated index count for this table
  long long  final_pos;
  int        final_val;
};

__device__ __forceinline__ void async_load_b128(unsigned lds_byte, const void* gaddr) {
  // memory -> LDS, 16B per lane, ASYNCcnt-tracked, non-temporal
  asm volatile("global_load_async_to_lds_b128 %0, %1, off th:TH_LOAD_NT"
               :: "v"(lds_byte), "v"(gaddr)
               : "memory");
}

__device__ __forceinline__ void async_store_b128(void* gaddr, unsigned lds_byte) {
  // LDS -> memory, 16B per lane (16B-aligned path), non-temporal
  asm volatile("global_store_async_from_lds_b128 %0, %1, off th:TH_STORE_NT"
               :: "v"(gaddr), "v"(lds_byte)
               : "memory");
}

__device__ __forceinline__ void async_store_b32(void* gaddr, unsigned lds_byte) {
  // LDS -> memory, 4B per lane (realigning path), non-temporal
  asm volatile("global_store_async_from_lds_b32 %0, %1, off th:TH_STORE_NT"
               :: "v"(gaddr), "v"(lds_byte)
               : "memory");
}

__device__ __forceinline__ void wait_asynccnt0() {
  asm volatile("s_wait_asynccnt 0" ::: "memory");
}

__global__ void __launch_bounds__(TPB)
seg_copy_async(SegParams p, int* __restrict__ out) {
  __shared__ int lds[TILE_CHUNKS * 4];   // 8 KiB

  const int s = blockIdx.y;
  const int* __restrict__ src = p.src[s];
  const long long D = p.dst[s];
  const int M = p.cnt[s];
  const int C = M >> 2;       // full 16B source chunks
  const int tail = M & 3;     // trailing words (0 for the reference sizes)
  const int j = threadIdx.x;

  if (blockIdx.x == 0 && j < tail) {
    out[D + 4LL * C + j] = src[4LL * C + j];
  }

  // AS(3)->flat pointers carry the LDS byte offset in their low 32 bits.
  const unsigned lbase = (unsigned)(unsigned long long)(&lds[0]);
  const bool dst_aligned = ((D & 3) == 0);
  const int tiles = (C + TILE_CHUNKS - 1) / TILE_CHUNKS;

  for (int t = blockIdx.x; t < tiles; t += gridDim.x) {
    const int c0 = t * TILE_CHUNKS;
    int nck = C - c0;
    if (nck > TILE_CHUNKS) nck = TILE_CHUNKS;

    // Phase 1: aligned wide async loads, memory -> LDS (no VGPR data).
#pragma unroll
    for (int h = 0; h < 2; ++h) {
      const int c = h * TPB + j;
      if (c < nck) {
        async_load_b128(lbase + 16u * (unsigned)c, src + 4LL * (c0 + c));
      }
    }
    wait_asynccnt0();                    // this wave's loads landed in LDS

    if (dst_aligned) {
      // Each thread stores exactly the chunks it loaded: wave-local LDS
      // dependencies only -> no block barrier needed.
#pragma unroll
      for (int h = 0; h < 2; ++h) {
        const int c = h * TPB + j;
        if (c < nck) {
          async_store_b128(out + D + 4LL * (c0 + c), lbase + 16u * (unsigned)c);
        }
      }
      wait_asynccnt0();                  // stores drained before LDS reuse
    } else {
      __syncthreads();                   // cross-wave LDS visibility
      const int tw = nck << 2;
      int* dbase = out + D + 4LL * c0;
#pragma unroll
      for (int k = 0; k < 8; ++k) {
        const int idx = k * TPB + j;     // contiguous across the block per pass
        if (idx < tw) {
          async_store_b32(dbase + idx, lbase + 4u * (unsigned)idx);
        }
      }
      wait_asynccnt0();                  // stores consumed LDS before overwrite
      __syncthreads();
    }
  }
}

__global__ void __launch_bounds__(TPB)
offsets_rebase(OffParams p, int* __restrict__ out) {
  const int t = blockIdx.y;
  const int i = blockIdx.x * blockDim.x + threadIdx.x;
  if (i < p.bcnt[t]) {
    out[p.dst[t] + i] = p.off[t][i] + p.rebase[t];
  }
  if (t == 0 && blockIdx.x == 0 && i == 0) {
    out[p.final_pos] = p.final_val;
  }
}

extern "C" void kernel_launch(void* const* d_in, const int* in_sizes, int n_in,
                              void* d_out, int out_size, void* d_ws, size_t ws_size,
                              hipStream_t stream) {
  (void)n_in; (void)out_size; (void)d_ws; (void)ws_size;

  // Input order: indices_0..7, offsets_0..7, weights_0..7
  long long accI[9]; accI[0] = 0;
  for (int i = 0; i < 8; ++i) accI[i + 1] = accI[i] + in_sizes[i];
  const long long IDX_TOTAL = accI[8];

  long long offAcc[9]; offAcc[0] = 0;
  for (int i = 0; i < 8; ++i) offAcc[i + 1] = offAcc[i] + (in_sizes[8 + i] - 1);
  const long long OFF_TOTAL = offAcc[8] + 1;   // + final grand-total element

  long long accW[9]; accW[0] = 0;
  for (int i = 0; i < 8; ++i) accW[i + 1] = accW[i] + in_sizes[16 + i];

  SegParams sp;
  for (int i = 0; i < 8; ++i) {
    sp.src[i] = (const int*)d_in[i];
    sp.dst[i] = accI[i];                               // 16B-aligned for ref sizes
    sp.cnt[i] = in_sizes[i];
  }
  for (int i = 0; i < 8; ++i) {
    sp.src[8 + i] = (const int*)d_in[16 + i];          // weights, bit-copied as i32
    sp.dst[8 + i] = IDX_TOTAL + OFF_TOTAL + accW[i];   // odd word offset -> dword path
    sp.cnt[8 + i] = in_sizes[16 + i];
  }

  OffParams op;
  int maxB = 1;
  for (int i = 0; i < 8; ++i) {
    op.off[i]    = (const int*)d_in[8 + i];
    op.dst[i]    = IDX_TOTAL + offAcc[i];
    op.bcnt[i]   = in_sizes[8 + i] - 1;
    op.rebase[i] = (int)accI[i];
    if (op.bcnt[i] > maxB) maxB = op.bcnt[i];
  }
  op.final_pos = IDX_TOTAL + OFF_TOTAL - 1;
  op.final_val = (int)IDX_TOTAL;

  int* out = (int*)d_out;

  // 16 segments x 256 blocks; each block streams 8 KiB tiles through LDS.
  seg_copy_async<<<dim3(256, 16, 1), TPB, 0, stream>>>(sp, out);

  const int bx = (maxB + 255) / 256;
  offsets_rebase<<<dim3(bx, 8, 1), TPB, 0, stream>>>(op, out);
}